// GatedResidualConv_28432683499607
// MI455X (gfx1250) — compile-verified
//
#include <hip/hip_runtime.h>
#include <math.h>

// ---------------------------------------------------------------------------
// GatedResidualConv for MI455X (gfx1250), fp32 WMMA path.
// g = v (q^T k) is rewritten as (v q^T) k  -> 17 GFLOP collapses to ~0.8 GFLOP
// and the [B,L,L] intermediate disappears; everything stays L2-resident.
// ---------------------------------------------------------------------------

typedef __attribute__((ext_vector_type(2))) float v2f;
typedef __attribute__((ext_vector_type(8))) float v8f;

#define NB 4
#define NC 64
#define NL 4096
#define LT 64                 // L columns per workgroup
#define NTILES (NL / LT)      // 64 tiles per batch
#define NWAVES 4
#define BLOCK (NWAVES * 32)
#define EPS_BN 1e-5f

#define SE_STRIDE 68          // 66 halo cols + pad (bank-conflict avoidance)
#define SH_STRIDE 65          // 64 cols + pad

__device__ __forceinline__ v8f wmma_f32(v2f a, v2f b, v8f c) {
  // V_WMMA_F32_16X16X4_F32: D(16x16,f32) = A(16x4,f32) * B(4x16,f32) + C
  return __builtin_amdgcn_wmma_f32_16x16x4_f32(false, a, false, b,
                                               (short)0, c, false, false);
}

// ---------------------------------------------------------------------------
// K1: per (batch, 64-col tile):
//   sE = elu(x) with +-1 circular halo (LDS)
//   h  = sum_t conv_w[:,:,t] @ sE_shift(t) + conv_b        (WMMA, K=64 x3)
//   q,k,v = W{q,k,v} @ h + b                               (WMMA)
//   k -> global kbuf ; Mpart(block) = v @ q^T  (64x64)     (WMMA, K=64)
// Each wave owns output-row band mt = wave (rows wave*16 .. wave*16+15).
// ---------------------------------------------------------------------------
__global__ __launch_bounds__(BLOCK)
void k1_conv_qkv_M(const float* __restrict__ x,
                   const float* __restrict__ conv_w, const float* __restrict__ conv_b,
                   const float* __restrict__ wq, const float* __restrict__ bq,
                   const float* __restrict__ wk, const float* __restrict__ bk,
                   const float* __restrict__ wv, const float* __restrict__ bv,
                   float* __restrict__ kbuf, float* __restrict__ Mpart) {
  __shared__ float sE[NC * SE_STRIDE];
  __shared__ float sH[NC * SH_STRIDE];
  __shared__ float sQ[NC * SH_STRIDE];
  __shared__ float sV[NC * SH_STRIDE];

  const int tid  = threadIdx.x;
  const int wave = tid >> 5;
  const int lane = tid & 31;
  const int half = lane >> 4;
  const int mn   = lane & 15;

  const int bIdx = blockIdx.x / NTILES;
  const int l0   = (blockIdx.x % NTILES) * LT;

  // ---- load x tile with circular halo, apply ELU --------------------------
  const float* xb = x + bIdx * NC * NL;
  for (int idx = tid; idx < NC * (LT + 2); idx += BLOCK) {
    int c = idx / (LT + 2);
    int j = idx % (LT + 2);
    int gl = (l0 + j - 1) & (NL - 1);
    float v = xb[c * NL + gl];
    sE[c * SE_STRIDE + j] = (v > 0.0f) ? v : (__expf(v) - 1.0f);
  }
  __syncthreads();

  const int mt = wave;                  // row band of this wave
  const int arow = mt * 16 + mn;        // A-fragment row
  const int rbase = mt * 16 + half * 8; // C/D fragment base row

  // ---- circular conv as 3 shifted K=64 GEMMs ------------------------------
  for (int nt = 0; nt < 4; ++nt) {
    const int bcol = nt * 16 + mn;
    v8f acc = {};
    for (int tap = 0; tap < 3; ++tap) {
      const float* Wt = conv_w + tap;   // A[m][k] = conv_w[(m*64+k)*3 + tap]
      for (int kk = 0; kk < 16; ++kk) {
        const int k0 = kk * 4 + half * 2;
        v2f a, b;
        a[0] = Wt[(arow * NC + k0) * 3];
        a[1] = Wt[(arow * NC + k0 + 1) * 3];
        b[0] = sE[k0 * SE_STRIDE + bcol + tap];
        b[1] = sE[(k0 + 1) * SE_STRIDE + bcol + tap];
        acc = wmma_f32(a, b, acc);
      }
    }
#pragma unroll
    for (int r = 0; r < 8; ++r)
      sH[(rbase + r) * SH_STRIDE + bcol] = acc[r] + conv_b[rbase + r];
  }
  __syncthreads();

  // ---- q, k, v = W @ h + bias ---------------------------------------------
  for (int nt = 0; nt < 4; ++nt) {
    const int bcol = nt * 16 + mn;
    v8f aq = {}, ak = {}, av = {};
    for (int kk = 0; kk < 16; ++kk) {
      const int k0 = kk * 4 + half * 2;
      v2f bf, a;
      bf[0] = sH[k0 * SH_STRIDE + bcol];
      bf[1] = sH[(k0 + 1) * SH_STRIDE + bcol];
      a[0] = wq[arow * NC + k0]; a[1] = wq[arow * NC + k0 + 1];
      aq = wmma_f32(a, bf, aq);
      a[0] = wk[arow * NC + k0]; a[1] = wk[arow * NC + k0 + 1];
      ak = wmma_f32(a, bf, ak);
      a[0] = wv[arow * NC + k0]; a[1] = wv[arow * NC + k0 + 1];
      av = wmma_f32(a, bf, av);
    }
#pragma unroll
    for (int r = 0; r < 8; ++r) {
      const int row = rbase + r;
      sQ[row * SH_STRIDE + bcol] = aq[r] + bq[row];
      sV[row * SH_STRIDE + bcol] = av[r] + bv[row];
      kbuf[(bIdx * NC + row) * NL + l0 + bcol] = ak[r] + bk[row];
    }
  }
  __syncthreads();

  // ---- partial M = v @ q^T over this 64-col slice -------------------------
  float* Mout = Mpart + blockIdx.x * (NC * NC);
  for (int nt = 0; nt < 4; ++nt) {
    const int bcol = nt * 16 + mn;
    v8f acc = {};
    for (int kk = 0; kk < 16; ++kk) {
      const int k0 = kk * 4 + half * 2;
      v2f a, b;
      a[0] = sV[arow * SH_STRIDE + k0];
      a[1] = sV[arow * SH_STRIDE + k0 + 1];
      b[0] = sQ[bcol * SH_STRIDE + k0];      // B[k][n] = q[n][l0+k] (transposed)
      b[1] = sQ[bcol * SH_STRIDE + k0 + 1];
      acc = wmma_f32(a, b, acc);
    }
#pragma unroll
    for (int r = 0; r < 8; ++r)
      Mout[(rbase + r) * NC + bcol] = acc[r];
  }
}

// K1b: deterministic reduction of the 64 per-tile Mpart slabs -> M[b] (64x64)
__global__ void k1b_reduceM(const float* __restrict__ Mpart,
                            float* __restrict__ Mbuf) {
  int i = blockIdx.x * blockDim.x + threadIdx.x;
  if (i >= NB * NC * NC) return;
  const int b = i >> 12;            // / 4096
  const int mn_ = i & 4095;
  const float* p = Mpart + (size_t)(b * NTILES) * (NC * NC) + mn_;
  float s = 0.0f;
  for (int t = 0; t < NTILES; ++t) s += p[(size_t)t * NC * NC];
  Mbuf[i] = s;
}

// ---------------------------------------------------------------------------
// K2: g = M[b] @ k (WMMA), y = g + x -> d_out; deterministic per-channel
// sum / sum-of-squares partials per block (each wave owns 16 channel rows).
// ---------------------------------------------------------------------------
__global__ __launch_bounds__(BLOCK)
void k2_gate_residual(const float* __restrict__ x,
                      const float* __restrict__ kbuf,
                      const float* __restrict__ Mbuf,
                      float* __restrict__ y,
                      float* __restrict__ sumPart, float* __restrict__ sqPart) {
  __shared__ float sK[NC * SH_STRIDE];
  __shared__ float sSum[NC * 16];
  __shared__ float sSq[NC * 16];

  const int tid  = threadIdx.x;
  const int wave = tid >> 5;
  const int lane = tid & 31;
  const int half = lane >> 4;
  const int mn   = lane & 15;

  const int bIdx = blockIdx.x / NTILES;
  const int l0   = (blockIdx.x % NTILES) * LT;

  const float* kb = kbuf + (size_t)bIdx * NC * NL + l0;
  for (int idx = tid; idx < NC * LT; idx += BLOCK) {
    int c = idx >> 6, j = idx & 63;
    sK[c * SH_STRIDE + j] = kb[c * NL + j];
  }
  __syncthreads();

  const float* Mb = Mbuf + bIdx * NC * NC;
  const float* xb = x + (size_t)bIdx * NC * NL + l0;
  float* yb = y + (size_t)bIdx * NC * NL + l0;

  const int mt = wave;
  const int arow = mt * 16 + mn;
  const int rbase = mt * 16 + half * 8;

  float regS[8], regQ[8];
#pragma unroll
  for (int r = 0; r < 8; ++r) { regS[r] = 0.0f; regQ[r] = 0.0f; }

  for (int nt = 0; nt < 4; ++nt) {
    const int bcol = nt * 16 + mn;
    v8f acc = {};
    for (int kk = 0; kk < 16; ++kk) {
      const int k0 = kk * 4 + half * 2;
      v2f a, b;
      a[0] = Mb[arow * NC + k0];
      a[1] = Mb[arow * NC + k0 + 1];
      b[0] = sK[k0 * SH_STRIDE + bcol];
      b[1] = sK[(k0 + 1) * SH_STRIDE + bcol];
      acc = wmma_f32(a, b, acc);
    }
#pragma unroll
    for (int r = 0; r < 8; ++r) {
      const int row = rbase + r;
      float yv = acc[r] + xb[row * NL + bcol];
      yb[row * NL + bcol] = yv;
      regS[r] += yv;
      regQ[r] += yv * yv;
    }
  }
  // rows are (wave, half)-exclusive: each (row, mn) cell written by one lane
#pragma unroll
  for (int r = 0; r < 8; ++r) {
    sSum[(rbase + r) * 16 + mn] = regS[r];
    sSq [(rbase + r) * 16 + mn] = regQ[r];
  }
  __syncthreads();
  if (tid < NC) {
    float s = 0.0f, q = 0.0f;
    for (int j = 0; j < 16; ++j) { s += sSum[tid * 16 + j]; q += sSq[tid * 16 + j]; }
    sumPart[blockIdx.x * NC + tid] = s;
    sqPart [blockIdx.x * NC + tid] = q;
  }
}

// K2b: fold per-block partials -> per-channel affine (gamma*rsqrt, shifted beta)
__global__ void k2b_stats(const float* __restrict__ sumPart,
                          const float* __restrict__ sqPart,
                          const float* __restrict__ gamma,
                          const float* __restrict__ beta,
                          float* __restrict__ scaleA, float* __restrict__ scaleB) {
  const int c = threadIdx.x;
  if (c >= NC) return;
  float s = 0.0f, q = 0.0f;
  for (int blk = 0; blk < NB * NTILES; ++blk) {
    s += sumPart[blk * NC + c];
    q += sqPart [blk * NC + c];
  }
  const float invN = 1.0f / (float)(NB * NL);
  const float mean = s * invN;
  const float var  = q * invN - mean * mean;   // biased variance (training BN)
  const float inv  = rsqrtf(var + EPS_BN);
  const float ga   = gamma[c] * inv;
  scaleA[c] = ga;
  scaleB[c] = beta[c] - mean * ga;
}

// K3: in-place vectorized normalize of d_out
__global__ void k3_bn(float* __restrict__ y,
                      const float* __restrict__ scaleA,
                      const float* __restrict__ scaleB) {
  int i4 = blockIdx.x * blockDim.x + threadIdx.x;   // float4 index
  if (i4 >= NB * NC * NL / 4) return;
  const int c = (i4 >> 10) & (NC - 1);              // NL/4 = 1024 per channel row
  float4 v = ((float4*)y)[i4];
  const float a = scaleA[c], b = scaleB[c];
  v.x = v.x * a + b;  v.y = v.y * a + b;
  v.z = v.z * a + b;  v.w = v.w * a + b;
  ((float4*)y)[i4] = v;
}

extern "C" void kernel_launch(void* const* d_in, const int* in_sizes, int n_in,
                              void* d_out, int out_size, void* d_ws, size_t ws_size,
                              hipStream_t stream) {
  (void)in_sizes; (void)n_in; (void)out_size; (void)ws_size;
  const float* x      = (const float*)d_in[0];
  const float* conv_w = (const float*)d_in[1];
  const float* conv_b = (const float*)d_in[2];
  const float* wq     = (const float*)d_in[3];
  const float* bq     = (const float*)d_in[4];
  const float* wk     = (const float*)d_in[5];
  const float* bk     = (const float*)d_in[6];
  const float* wv     = (const float*)d_in[7];
  const float* bv     = (const float*)d_in[8];
  const float* gamma  = (const float*)d_in[9];
  const float* beta   = (const float*)d_in[10];
  float* out = (float*)d_out;

  // workspace layout (floats)
  float* ws      = (float*)d_ws;
  float* kbuf    = ws;                                   // NB*NC*NL   = 1048576
  float* Mpart   = kbuf    + (size_t)NB * NC * NL;       // 256*4096   = 1048576
  float* Mbuf    = Mpart   + (size_t)NB * NTILES * NC * NC;  // NB*4096
  float* sumPart = Mbuf    + (size_t)NB * NC * NC;       // 256*64
  float* sqPart  = sumPart + (size_t)NB * NTILES * NC;   // 256*64
  float* scaleA  = sqPart  + (size_t)NB * NTILES * NC;   // 64
  float* scaleB  = scaleA  + NC;                         // 64

  const int grid1 = NB * NTILES;  // 256 workgroups
  k1_conv_qkv_M<<<grid1, BLOCK, 0, stream>>>(x, conv_w, conv_b, wq, bq, wk, bk,
                                             wv, bv, kbuf, Mpart);
  k1b_reduceM<<<(NB * NC * NC + 255) / 256, 256, 0, stream>>>(Mpart, Mbuf);
  k2_gate_residual<<<grid1, BLOCK, 0, stream>>>(x, kbuf, Mbuf, out, sumPart, sqPart);
  k2b_stats<<<1, NC, 0, stream>>>(sumPart, sqPart, gamma, beta, scaleA, scaleB);
  k3_bn<<<(NB * NC * NL / 4 + 255) / 256, 256, 0, stream>>>(out, scaleA, scaleB);
}